// PaniniGenomicGrammar_78391743086720
// MI455X (gfx1250) — compile-verified
//
#include <hip/hip_runtime.h>
#include <hip/hip_bf16.h>

typedef __attribute__((ext_vector_type(2))) float v2f;
typedef __attribute__((ext_vector_type(4))) float v4f;
typedef __attribute__((ext_vector_type(8))) float v8f;

// Problem constants (from reference): hidden [2,4096,1024], A [16,1024,8], B [8,32000]
#define NROWS   8192      // 2*4096 flattened (b,t) rows
#define TLEN    4096
#define EDIM    1024
#define RDIM    8
#define VDIM    32000
#define BUCKET_SHIFT 8    // t / 256

// ---------------------------------------------------------------------------
// Stage 1: h_proj[row, r] = alpha * sum_e hidden[row, e] * A[bucket(row), e, r]
// One wave32 per row. Lane-strided over E, 8 accumulators, butterfly reduce.
// Traffic: 33.5 MB hidden from HBM; A (512 KB) stays in L2. ~1.5 us.
// ---------------------------------------------------------------------------
__global__ void __launch_bounds__(256)
stage1_proj(const float* __restrict__ hidden,
            const float* __restrict__ A,
            const float* __restrict__ alpha_p,
            float* __restrict__ hproj) {
    const int row  = blockIdx.x * 8 + (threadIdx.x >> 5);
    const int lane = threadIdx.x & 31;
    const int t      = row & (TLEN - 1);
    const int bucket = t >> BUCKET_SHIFT;           // 0..15, already clamped by t<4096

    const float* __restrict__ hrow = hidden + (size_t)row * EDIM;
    const float* __restrict__ Ab   = A + (size_t)bucket * EDIM * RDIM;

    float acc[RDIM];
#pragma unroll
    for (int r = 0; r < RDIM; ++r) acc[r] = 0.0f;

#pragma unroll 4
    for (int e = lane; e < EDIM; e += 32) {
        const float h = hrow[e];
        const v4f a0 = *(const v4f*)(Ab + (size_t)e * RDIM);
        const v4f a1 = *(const v4f*)(Ab + (size_t)e * RDIM + 4);
        acc[0] += h * a0.x;  acc[1] += h * a0.y;
        acc[2] += h * a0.z;  acc[3] += h * a0.w;
        acc[4] += h * a1.x;  acc[5] += h * a1.y;
        acc[6] += h * a1.z;  acc[7] += h * a1.w;
    }

    // wave32 butterfly reduction (warpSize == 32 on gfx1250)
#pragma unroll
    for (int off = 16; off > 0; off >>= 1) {
#pragma unroll
        for (int r = 0; r < RDIM; ++r)
            acc[r] += __shfl_xor(acc[r], off, 32);
    }

    const float alpha = *alpha_p;
    if (lane < RDIM)
        hproj[(size_t)row * RDIM + lane] = alpha * acc[lane];
}

// ---------------------------------------------------------------------------
// Stage 2: out[row, v] = sum_r hproj[row, r] * B[r, v]   (alpha already folded)
// Rank-8 GEMM, 1.048 GB of output -> pure store-bandwidth bound (~46 us floor).
// Each wave owns one 16-wide vocab tile, keeps its B fragments in registers,
// and streams 8 consecutive 16-row tiles using V_WMMA_F32_16X16X4_F32 (x2 for
// K=8), preserving full fp32 precision like the reference.
//
// f32 16x16x4 WMMA layouts (ISA 7.12.2):
//   A (16x4): lanes 0-15 -> M=lane, {V0,V1}={K0,K1}; lanes 16-31 -> {K2,K3}
//   B (4x16): lanes 0-15 -> N=lane, {V0,V1}={K0,K1}; lanes 16-31 -> {K2,K3}
//   C/D     : VGPR j, lanes 0-15 -> M=j,  N=lane
//                      lanes 16-31 -> M=j+8, N=lane-16
// ---------------------------------------------------------------------------
__global__ void __launch_bounds__(256)
stage2_expand(const float* __restrict__ hproj,
              const float* __restrict__ Bm,
              float* __restrict__ out) {
    const int lane   = threadIdx.x & 31;
    const int waveid = threadIdx.x >> 5;
    const int n      = lane & 15;          // column-in-tile (B/C/D), row-in-tile (A)
    const int khalf  = lane >> 4;          // 0: K{0,1}, 1: K{2,3} (per wmma)

    const int v0 = (blockIdx.x * 8 + waveid) * 16;     // vocab tile base

    // B fragments: wmma0 covers r=0..3, wmma1 covers r=4..7 (loop-invariant)
    v2f bf0, bf1;
    bf0.x = Bm[(size_t)(2 * khalf + 0) * VDIM + v0 + n];
    bf0.y = Bm[(size_t)(2 * khalf + 1) * VDIM + v0 + n];
    bf1.x = Bm[(size_t)(4 + 2 * khalf + 0) * VDIM + v0 + n];
    bf1.y = Bm[(size_t)(4 + 2 * khalf + 1) * VDIM + v0 + n];

    const int mbase = khalf * 8;           // C/D row offset for hi lanes

#pragma unroll 1
    for (int i = 0; i < 8; ++i) {
        const int row0 = (blockIdx.y * 8 + i) * 16;

        // A fragments from h_proj: lane's row = row0 + n, 32B row -> two b64 loads
        const float* __restrict__ hr = hproj + (size_t)(row0 + n) * RDIM;
        const v2f af0 = *(const v2f*)(hr + 2 * khalf);       // r = 2k, 2k+1
        const v2f af1 = *(const v2f*)(hr + 4 + 2 * khalf);   // r = 4+2k, 4+2k+1

        v8f c = {};
        c = __builtin_amdgcn_wmma_f32_16x16x4_f32(false, af0, false, bf0,
                                                  (short)0, c, false, false);
        c = __builtin_amdgcn_wmma_f32_16x16x4_f32(false, af1, false, bf1,
                                                  (short)0, c, false, false);

        // Scatter the 16x16 f32 tile: 8 b32 stores, each covering two 64B runs
        float* __restrict__ ob = out + (size_t)row0 * VDIM + v0 + n;
#pragma unroll
        for (int j = 0; j < 8; ++j)
            ob[(size_t)(mbase + j) * VDIM] = c[j];
    }
}

// ---------------------------------------------------------------------------
extern "C" void kernel_launch(void* const* d_in, const int* in_sizes, int n_in,
                              void* d_out, int out_size, void* d_ws, size_t ws_size,
                              hipStream_t stream) {
    const float* hidden  = (const float*)d_in[0];   // [2,4096,1024]
    const float* A       = (const float*)d_in[1];   // [16,1024,8]
    const float* Bm      = (const float*)d_in[2];   // [8,32000]
    const float* alpha_p = (const float*)d_in[3];   // scalar
    float*       out     = (float*)d_out;           // [2,4096,32000]
    float*       hproj   = (float*)d_ws;            // [8192,8] = 256 KB scratch

    (void)in_sizes; (void)n_in; (void)out_size; (void)ws_size;

    // Stage 1: 8192 rows, 8 waves/block -> 1024 blocks
    stage1_proj<<<dim3(NROWS / 8), dim3(256), 0, stream>>>(hidden, A, alpha_p, hproj);

    // Stage 2: 2000 v-tiles / 8 waves = 250 blocks in x; 512 row-tiles / 8 per wave
    //          = 64 blocks in y.
    stage2_expand<<<dim3(250, 64), dim3(256), 0, stream>>>(hproj, Bm, out);
}